// Attention_Decoder_10797547782159
// MI455X (gfx1250) — compile-verified
//
#include <hip/hip_runtime.h>

// ---------------------------------------------------------------------------
// Attention decoder (B=128, T=128, U=64, M=64) for MI455X / gfx1250.
//   Kernel 1: r2 = h_en_all @ Ue  as f32 WMMA GEMM (V_WMMA_F32_16X16X4_F32)
//   Kernel 2: persistent per-batch workgroup runs the 127-step scan with
//             r2[b], h_en[b], Wr resident in LDS (staged via async-to-LDS),
//             wave32 shuffle reductions to minimize barriers.
// ---------------------------------------------------------------------------

typedef __attribute__((ext_vector_type(2))) float v2f;
typedef __attribute__((ext_vector_type(8))) float v8f;

#define B_ 128
#define T_ 128
#define U_ 64
#define M_ 64

__device__ __forceinline__ float fast_tanh(float x) {
#if __has_builtin(__builtin_amdgcn_tanhf)
  return __builtin_amdgcn_tanhf(x);
#else
  return tanhf(x);
#endif
}

__device__ __forceinline__ float sigm(float x) {
  return 1.0f / (1.0f + expf(-x));
}

__device__ __forceinline__ float wred_sum(float v) {
#pragma unroll
  for (int o = 16; o > 0; o >>= 1) v += __shfl_xor(v, o, 32);
  return v;
}

__device__ __forceinline__ float wred_max(float v) {
#pragma unroll
  for (int o = 16; o > 0; o >>= 1) v = fmaxf(v, __shfl_xor(v, o, 32));
  return v;
}

__device__ __forceinline__ void async_b128(const float* g, const float* lds) {
  unsigned la = (unsigned)(unsigned long long)(const void*)lds;
  unsigned long long ga = (unsigned long long)(const void*)g;
  asm volatile("global_load_async_to_lds_b128 %0, %1, off"
               :
               : "v"(la), "v"(ga)
               : "memory");
}

// ---------------------------------------------------------------------------
// Kernel 1: r2[(b,t), u] = sum_m hen[(b,t), m] * Ue[m, u]
// GEMM (16384 x 64) x (64 x 64), fp32, via WMMA 16x16x4 f32.
// One wave computes a 16-row stripe across all 64 output columns
// (4 accumulators, 16 K-steps, 64 wmma ops per wave). 1024 waves total.
// ---------------------------------------------------------------------------
__global__ __launch_bounds__(256) void r2_gemm_wmma(
    const float* __restrict__ hen, const float* __restrict__ Ue,
    float* __restrict__ r2) {
  const int lane = threadIdx.x & 31;
  const int gwave = (blockIdx.x * blockDim.x + threadIdx.x) >> 5;
  const int rowBase = gwave * 16;  // 16384/16 = 1024 row tiles
  // A (16x4 f32) layout: lanes 0-15 -> rows, K0/K1; lanes 16-31 -> rows, K2/K3
  const int r = rowBase + (lane & 15);
  const int khalf = (lane >> 4) << 1;  // 0 or 2
  const int nlane = lane & 15;

  v8f acc[4] = {};
#pragma unroll
  for (int k = 0; k < M_; k += 4) {
    const int kk = k + khalf;
    v2f a = *(const v2f*)(hen + r * M_ + kk);  // K=kk, K=kk+1 (8B aligned)
#pragma unroll
    for (int tn = 0; tn < 4; ++tn) {
      const int n = tn * 16 + nlane;
      v2f bb;
      bb.x = Ue[(kk + 0) * U_ + n];
      bb.y = Ue[(kk + 1) * U_ + n];
      acc[tn] = __builtin_amdgcn_wmma_f32_16x16x4_f32(
          /*neg_a=*/false, a, /*neg_b=*/false, bb,
          /*c_mod=*/(short)0, acc[tn], /*reuse_a=*/false, /*reuse_b=*/false);
    }
  }
  // C/D layout: VGPR j, lanes 0-15 -> row rowBase+j; lanes 16-31 -> rowBase+8+j
#pragma unroll
  for (int tn = 0; tn < 4; ++tn) {
    const int col = tn * 16 + nlane;
#pragma unroll
    for (int j = 0; j < 8; ++j) {
      const int row = rowBase + j + ((lane >> 4) << 3);
      r2[row * U_ + col] = acc[tn][j];
    }
  }
}

// ---------------------------------------------------------------------------
// Kernel 2: one workgroup (128 threads = 4 waves) per batch element b.
// ---------------------------------------------------------------------------
__global__ __launch_bounds__(128) void attn_decoder_scan(
    const float* __restrict__ hen_g, const float* __restrict__ r2_g,
    const float* __restrict__ y_g, const float* __restrict__ h0,
    const float* __restrict__ c0, const float* __restrict__ We,
    const float* __restrict__ ve_g, const float* __restrict__ Wp,
    const float* __restrict__ bp, const float* __restrict__ Wk,
    const float* __restrict__ Wr_g, const float* __restrict__ bias,
    float* __restrict__ out) {
  __shared__ float s_r2[T_ * U_];        // 32 KB, [t*64+u]
  __shared__ float s_hen[T_ * M_];       // 32 KB, [t*64+m]
  __shared__ float s_Wr[U_ * 4 * U_];    // 64 KB, [u*256+j]
  __shared__ float s_h[U_], s_c[U_];
  __shared__ float s_q[U_];
  __shared__ float s_beta[T_];
  __shared__ float s_ctx[M_];
  __shared__ float s_red[T_];            // q/ctx partial sums
  __shared__ float s_z[4 * U_];
  __shared__ float s_ve[U_], s_Wk[4 * U_], s_bias[4 * U_], s_Wp[1 + M_];
  __shared__ float s_xp[16];             // cross-wave scalar slots
  __shared__ float s_x, s_bp;

  const int tid = threadIdx.x;
  const int wv = tid >> 5;
  const int lane = tid & 31;
  const int b = blockIdx.x;

  // ---- Stage r2[b], hen[b], Wr into LDS with async-to-LDS (b128 chunks) ----
  const float* g_r2 = r2_g + (size_t)b * T_ * U_;
  const float* g_hen = hen_g + (size_t)b * T_ * M_;
  for (int i = tid * 4; i < T_ * U_; i += 128 * 4) {
    async_b128(g_r2 + i, &s_r2[i]);
    async_b128(g_hen + i, &s_hen[i]);
  }
  for (int i = tid * 4; i < U_ * 4 * U_; i += 128 * 4)
    async_b128(Wr_g + i, &s_Wr[i]);
  // ---- Small operands via normal loads ----
  if (tid < U_) {
    s_h[tid] = h0[b * U_ + tid];
    s_c[tid] = c0[b * U_ + tid];
    s_ve[tid] = ve_g[tid];
  }
  for (int i = tid; i < 4 * U_; i += 128) {
    s_Wk[i] = Wk[i];
    s_bias[i] = bias[i];
  }
  if (tid < 1 + M_) s_Wp[tid] = Wp[tid];
  if (tid == 0) s_bp = bp[0];
  asm volatile("s_wait_asynccnt 0" ::: "memory");
  __syncthreads();

  const float* gy = y_g + (size_t)b * T_;  // y is (B,T,1)
  const int u64 = tid & 63;
  const int half = tid >> 6;

  for (int step = 0; step < T_; ++step) {
    // ---- q[u] = sum_k [h;c][k] * We[k,u], k-range split over half-groups ----
    {
      const float* hsrc = half ? s_c : s_h;  // half 0: k=0..63, half 1: 64..127
      float p = 0.f;
#pragma unroll 8
      for (int kk = 0; kk < 64; ++kk)
        p += hsrc[kk] * We[(half * 64 + kk) * U_ + u64];
      s_red[tid] = p;
      __syncthreads();  // (1)
      if (tid < U_) s_q[tid] = s_red[tid] + s_red[tid + 64];
      __syncthreads();  // (2)
    }

    // ---- e[t] = sum_u tanh(q[u] + r2[t,u]) * ve[u] (rotated u: bank spread)
    float e = 0.f;
    {
      const int t = tid;
#pragma unroll 8
      for (int i = 0; i < U_; ++i) {
        const int u = (t + i) & 63;
        e += fast_tanh(s_q[u] + s_r2[t * U_ + u]) * s_ve[u];
      }
    }

    // ---- softmax over t: wave-local shuffle reduce + 4-slot LDS combine ----
    {
      float wm = wred_max(e);
      if (lane == 0) s_xp[4 + wv] = wm;
      __syncthreads();  // (3)
      const float mx =
          fmaxf(fmaxf(s_xp[4], s_xp[5]), fmaxf(s_xp[6], s_xp[7]));
      const float ex = expf(e - mx);
      float ws = wred_sum(ex);
      if (lane == 0) s_xp[8 + wv] = ws;
      __syncthreads();  // (4)
      const float tot = (s_xp[8] + s_xp[9]) + (s_xp[10] + s_xp[11]);
      s_beta[tid] = ex / tot;
      __syncthreads();  // (5)
    }

    // ---- ctx[m] = sum_t beta[t] * hen[t,m] (bank = m: conflict free) ----
    {
      float p = 0.f;
#pragma unroll 8
      for (int t = half * 64; t < half * 64 + 64; ++t)
        p += s_beta[t] * s_hen[t * M_ + u64];
      s_red[tid] = p;
      __syncthreads();  // (6)
    }
    float cv = 0.f;
    if (tid < M_) {
      cv = s_red[tid] + s_red[tid + 64];
      s_ctx[tid] = cv;
    }

    if (step == T_ - 1) break;  // final attention only feeds the output ctx

    // ---- x = [y_t, ctx] @ Wp + bp (scalar), shuffle-reduced ----
    {
      float xpart = (tid < M_) ? cv * s_Wp[1 + tid] : 0.f;
      float ws = wred_sum(xpart);
      if (lane == 0) s_xp[wv] = ws;
      __syncthreads();  // (7) publishes s_ctx + partials
      if (tid == 0)
        s_x = gy[step] * s_Wp[0] +
              ((s_xp[0] + s_xp[1]) + (s_xp[2] + s_xp[3])) + s_bp;
      __syncthreads();  // (8)
    }

    // ---- z[j] = x*Wk[j] + h @ Wr[:,j] + b[j]; thread owns j and j+128 ----
    {
      const float x = s_x;
      const int j0 = tid, j1 = tid + 128;
      float z0 = x * s_Wk[j0] + s_bias[j0];
      float z1 = x * s_Wk[j1] + s_bias[j1];
#pragma unroll 8
      for (int u = 0; u < U_; ++u) {
        const float hv = s_h[u];
        z0 += hv * s_Wr[u * (4 * U_) + j0];
        z1 += hv * s_Wr[u * (4 * U_) + j1];
      }
      s_z[j0] = z0;
      s_z[j1] = z1;
    }
    __syncthreads();  // (9)

    // ---- LSTM cell update (keras gate order i, f, g, o) ----
    if (tid < U_) {
      const float zi = s_z[tid];
      const float zf = s_z[U_ + tid];
      const float zg = s_z[2 * U_ + tid];
      const float zo = s_z[3 * U_ + tid];
      const float cn = sigm(zf) * s_c[tid] + sigm(zi) * fast_tanh(zg);
      s_c[tid] = cn;
      s_h[tid] = sigm(zo) * fast_tanh(cn);
    }
    __syncthreads();  // (10)
  }

  // outputs: [h_f (B,U) | ctx (B,1,M)] flat (same thread wrote s_ctx[tid])
  if (tid < U_) {
    out[(size_t)b * U_ + tid] = s_h[tid];
    out[(size_t)B_ * U_ + (size_t)b * M_ + tid] = s_ctx[tid];
  }
}

extern "C" void kernel_launch(void* const* d_in, const int* in_sizes, int n_in,
                              void* d_out, int out_size, void* d_ws,
                              size_t ws_size, hipStream_t stream) {
  const float* hen = (const float*)d_in[0];    // (B,T,M)
  const float* y = (const float*)d_in[1];      // (B,T,1)
  const float* c0 = (const float*)d_in[2];     // "s" = cell state (B,U)
  const float* h0 = (const float*)d_in[3];     // h (B,U)
  const float* We = (const float*)d_in[4];     // (2U,U)
  const float* Ue = (const float*)d_in[5];     // (M,U)
  const float* ve = (const float*)d_in[6];     // (U,1)
  const float* Wp = (const float*)d_in[7];     // (1+M,1)
  const float* bp = (const float*)d_in[8];     // (1,)
  const float* Wk = (const float*)d_in[9];     // (1,4U)
  const float* Wr = (const float*)d_in[10];    // (U,4U)
  const float* bias = (const float*)d_in[11];  // (4U,)
  float* r2 = (float*)d_ws;                    // (B,T,U) = 4 MB scratch
  float* out = (float*)d_out;

  // 1024 waves (one per 16-row tile), 8 waves per 256-thread block.
  r2_gemm_wmma<<<(B_ * T_ / 16) / 8, 256, 0, stream>>>(hen, Ue, r2);
  // One persistent workgroup per batch element.
  attn_decoder_scan<<<B_, 128, 0, stream>>>(hen, r2, y, h0, c0, We, ve, Wp, bp,
                                            Wk, Wr, bias, out);
}